// HGNNLayer_65670049956246
// MI455X (gfx1250) — compile-verified
//
#include <hip/hip_runtime.h>

// ---------------------------------------------------------------------------
// HGNN layer for MI455X (gfx1250, wave32, WMMA).
//
// Reformulation: per-edge (feats @ A[type]) then scatter  ==  scatter of
// normalized source rows into 12 per-(type,slot) buffers P, then ONE dense
// GEMM  H = [X | P0..P11] @ [C_w^T ; A1[t] ; A2[t].top ; A2[t].bot] + C_b.
// GEMM runs on V_WMMA_F32_16X16X4_F32 (fp32, exact vs reference).
// All 13 weight blocks are pre-transposed into WT[b][n][k] so both A and B
// fragments are contiguous b64 loads in the hot loop.
// ---------------------------------------------------------------------------

#define N_NODES   200000
#define DIM       128
#define NUM_TYPES 4
#define NE1       400000
#define NE2       400000
#define NBLK      13            // 1 (X/Cw) + 4 (A1) + 8 (A2 halves)

typedef float v2f __attribute__((ext_vector_type(2)));
typedef float v8f __attribute__((ext_vector_type(8)));

// ---------------------------------------------------------------- zero ws
__global__ void zero_ws_kernel(float4* __restrict__ p, long n4) {
    long i = (long)blockIdx.x * blockDim.x + threadIdx.x;
    if (i < n4) p[i] = make_float4(0.f, 0.f, 0.f, 0.f);
}

// ------------------------------------------------- weight transpose (once)
// WT[b][n][k] = W_b[k][n]  so the GEMM B-fragment is a contiguous float2.
// Block 0: B[k][n] = Cw[n][k]  -> WT[0][n][k] = Cw[n][k] (direct copy).
__global__ void wt_kernel(const float* __restrict__ Cw,
                          const float* __restrict__ A1,
                          const float* __restrict__ A2,
                          float* __restrict__ WT) {
    int gid = blockIdx.x * blockDim.x + threadIdx.x;   // NBLK*128*128 threads
    if (gid >= NBLK * DIM * DIM) return;
    int b = gid >> 14;          // /16384
    int n = (gid >> 7) & 127;
    int k = gid & 127;
    float v;
    if (b == 0)      v = Cw[n * DIM + k];
    else if (b < 5)  v = A1[(long)(b - 1) * DIM * DIM + (long)k * DIM + n];
    else if (b < 9)  v = A2[(long)(b - 5) * 2 * DIM * DIM + (long)k * DIM + n];
    else             v = A2[(long)(b - 9) * 2 * DIM * DIM + (long)(k + DIM) * DIM + n];
    WT[gid] = v;
}

// ------------------------------------------------------------- edge counts
__global__ void count_kernel(const int* __restrict__ dest_idx, int dstride,
                             const int* __restrict__ types,
                             int* __restrict__ counts, int E) {
    int e = blockIdx.x * blockDim.x + threadIdx.x;
    if (e >= E) return;
    int dest = dest_idx[(long)e * dstride];
    int t    = types[e];
    atomicAdd(&counts[(long)t * N_NODES + dest], 1);
}

// ----------------------------------------------------- scatter shape-1 edges
// P[t][dest][:] += x[src][:] / count1[t][dest]; one edge per 32 lanes.
__global__ void scatter1_kernel(const float* __restrict__ x,
                                const int*  __restrict__ idx1,   // [2, E1]
                                const int*  __restrict__ type1,
                                const int*  __restrict__ counts, // [4, N]
                                float* __restrict__ P) {
    long gid  = (long)blockIdx.x * blockDim.x + threadIdx.x;
    int  e    = (int)(gid >> 5);
    int  lane = (int)(gid & 31);
    if (e >= NE1) return;
    int src  = idx1[e];
    int dest = idx1[NE1 + e];
    int t    = type1[e];
    float norm = 1.0f / (float)counts[(long)t * N_NODES + dest];
    const float4 v = *(const float4*)(x + (long)src * DIM + lane * 4);
    float* p = P + ((long)t * N_NODES + dest) * DIM + lane * 4;
    atomicAdd(p + 0, v.x * norm);
    atomicAdd(p + 1, v.y * norm);
    atomicAdd(p + 2, v.z * norm);
    atomicAdd(p + 3, v.w * norm);
}

// ----------------------------------------------------- scatter shape-2 edges
// Half-edge j: e=j>>1, which=j&1. Buffer 4+4*which+t += x[src]/count2[t][dest].
__global__ void scatter2_kernel(const float* __restrict__ x,
                                const int*  __restrict__ idx2,   // [2, 2*E2]
                                const int*  __restrict__ type2,
                                const int*  __restrict__ counts, // counts2 [4, N]
                                float* __restrict__ P) {
    long gid  = (long)blockIdx.x * blockDim.x + threadIdx.x;
    int  j    = (int)(gid >> 5);
    int  lane = (int)(gid & 31);
    if (j >= 2 * NE2) return;
    int e     = j >> 1;
    int which = j & 1;
    int src   = idx2[2 * e + which];
    int dest  = idx2[2 * NE2 + 2 * e];
    int t     = type2[e];
    float norm = 1.0f / (float)counts[(long)t * N_NODES + dest];
    const float4 v = *(const float4*)(x + (long)src * DIM + lane * 4);
    int buf = 4 + 4 * which + t;
    float* p = P + ((long)buf * N_NODES + dest) * DIM + lane * 4;
    atomicAdd(p + 0, v.x * norm);
    atomicAdd(p + 1, v.y * norm);
    atomicAdd(p + 2, v.z * norm);
    atomicAdd(p + 3, v.w * norm);
}

// --------------------------------------------------------------- fused GEMM
// H[N,128] = X @ C_w^T + sum_b P[b] @ W[b] + C_b
// One wave -> one 16x16 tile via v_wmma_f32_16x16x4_f32; 8 waves per block
// cover a 16-row slab across all 128 output columns. All A/B fragment loads
// are contiguous b64 (A row-major, B pre-transposed in WT).
__global__ __launch_bounds__(256) void gemm_kernel(
    const float* __restrict__ x,
    const float* __restrict__ P,
    const float* __restrict__ WT,   // [13,128,128]: WT[b][n][k] = W_b[k][n]
    const float* __restrict__ Cb,   // [128]
    float* __restrict__ out) {
    const int lane = threadIdx.x & 31;
    const int wave = threadIdx.x >> 5;
    const int l15  = lane & 15;
    const int h    = lane >> 4;          // lane half (0/1)
    const int m0   = blockIdx.x * 16;    // output row block
    const int n0   = wave * 16;          // output col block

    v8f acc = {};

    for (int blk = 0; blk < NBLK; ++blk) {
        const float* Arow = (blk == 0)
            ? x + (long)(m0 + l15) * DIM
            : P + ((long)(blk - 1) * N_NODES + m0 + l15) * DIM;
        const float* Brow = WT + (long)blk * DIM * DIM + (long)(n0 + l15) * DIM;
        #pragma unroll
        for (int k = 0; k < DIM; k += 4) {
            v2f a = *(const v2f*)(Arow + k + 2 * h);
            v2f b = *(const v2f*)(Brow + k + 2 * h);
            acc = __builtin_amdgcn_wmma_f32_16x16x4_f32(
                false, a, false, b, (short)0, acc, false, false);
        }
    }

    // Store D (M = r + 8h, N = n0 + l15) with bias.
    const float bias = Cb[n0 + l15];
    #pragma unroll
    for (int r = 0; r < 8; ++r) {
        out[(long)(m0 + r + 8 * h) * DIM + n0 + l15] = acc[r] + bias;
    }
}

// ---------------------------------------------------------------------------
extern "C" void kernel_launch(void* const* d_in, const int* in_sizes, int n_in,
                              void* d_out, int out_size, void* d_ws, size_t ws_size,
                              hipStream_t stream) {
    const float* x     = (const float*)d_in[0];
    const int*   idx1  = (const int*)  d_in[1];
    const int*   type1 = (const int*)  d_in[2];
    const int*   idx2  = (const int*)  d_in[3];
    const int*   type2 = (const int*)  d_in[4];
    const float* Cw    = (const float*)d_in[5];
    const float* Cb    = (const float*)d_in[6];
    const float* A1    = (const float*)d_in[7];
    const float* A2    = (const float*)d_in[8];
    float*       out   = (float*)d_out;

    // Workspace: P[12][N][128] f32 | counts1[4][N] + counts2[4][N] int | WT[13][128][128] f32.
    float* P       = (float*)d_ws;
    const long Pelems = (long)12 * N_NODES * DIM;            // 307.2M floats
    int*   counts1 = (int*)(P + Pelems);
    int*   counts2 = counts1 + (long)NUM_TYPES * N_NODES;
    float* WT      = (float*)(counts2 + (long)NUM_TYPES * N_NODES);
    const long zeroBytes = Pelems * 4 + (long)8 * N_NODES * 4;   // P + counts only
    const long n4 = zeroBytes / 16;

    // 1) zero P + counts (graph-capture-safe kernel, 16B stores)
    zero_ws_kernel<<<(unsigned)((n4 + 255) / 256), 256, 0, stream>>>((float4*)d_ws, n4);
    // 1b) build transposed weight blocks (independent of zeroing)
    wt_kernel<<<(NBLK * DIM * DIM + 255) / 256, 256, 0, stream>>>(Cw, A1, A2, WT);
    // 2) per-(type,dest) counts
    count_kernel<<<(NE1 + 255) / 256, 256, 0, stream>>>(idx1 + NE1, 1, type1, counts1, NE1);
    count_kernel<<<(NE2 + 255) / 256, 256, 0, stream>>>(idx2 + 2 * NE2, 2, type2, counts2, NE2);
    // 3) normalized scatter into pre-aggregation buffers
    {
        long threads = (long)NE1 * 32;
        scatter1_kernel<<<(unsigned)((threads + 255) / 256), 256, 0, stream>>>(
            x, idx1, type1, counts1, P);
    }
    {
        long threads = (long)2 * NE2 * 32;
        scatter2_kernel<<<(unsigned)((threads + 255) / 256), 256, 0, stream>>>(
            x, idx2, type2, counts2, P);
    }
    // 4) fused WMMA GEMM: [N,1664] @ [1664,128] + bias
    gemm_kernel<<<N_NODES / 16, 256, 0, stream>>>(x, P, WT, Cb, out);
}